// TropicalLinear_66116726555066
// MI455X (gfx1250) — compile-verified
//
#include <hip/hip_runtime.h>
#include <stdint.h>

// Tropical (max-plus) linear: y[b,o] = max_i (x[b,i] + W[o,i])
// B=1024, K=512, N=512, f32. VALU-bound => register/LDS tiling + v_max3.
// Data movement via CDNA5 async global->LDS copies (ASYNCcnt), double-buffered.

#define B_DIM 1024
#define K_DIM 512
#define N_DIM 512

constexpr int BM = 64;   // B-rows per block
constexpr int BN = 64;   // N-cols per block
constexpr int BK = 64;   // K-chunk
constexpr int TM = 4;    // per-thread B-rows
constexpr int TN = 4;    // per-thread N-cols
constexpr int NTHREADS = 256;
constexpr int NCHUNK = K_DIM / BK;  // 8

typedef __attribute__((address_space(3))) int lds_int_t;
typedef __attribute__((address_space(1))) int gbl_int_t;

__device__ __forceinline__ lds_int_t* to_lds(void* p) {
  return (lds_int_t*)(uint32_t)(uintptr_t)p;   // LDS addr = low 32 bits
}
__device__ __forceinline__ gbl_int_t* to_gbl(const void* p) {
  return (gbl_int_t*)(uintptr_t)p;
}

// One-dword async copy global -> LDS (per-lane independent LDS destination,
// which gives a free transpose while keeping global reads coalesced).
__device__ __forceinline__ void async_copy_b32(const float* g, float* l) {
#if __has_builtin(__builtin_amdgcn_global_load_async_to_lds_b32)
  __builtin_amdgcn_global_load_async_to_lds_b32(to_gbl(g), to_lds(l), 0, 0);
#else
  uint32_t lds_addr = (uint32_t)(uintptr_t)l;
  uint64_t gaddr = (uint64_t)(uintptr_t)g;
  asm volatile("global_load_async_to_lds_b32 %0, %1, off"
               :: "v"(lds_addr), "v"(gaddr) : "memory");
#endif
}

template <int N>
__device__ __forceinline__ void wait_async() {
#if __has_builtin(__builtin_amdgcn_s_wait_asynccnt)
  __builtin_amdgcn_s_wait_asynccnt(N);
#else
  asm volatile("s_wait_asynccnt %0" :: "i"(N) : "memory");
#endif
}

__global__ __launch_bounds__(NTHREADS)
void TropicalLinear_66116726555066_kernel(const float* __restrict__ x,
                                          const float* __restrict__ W,
                                          float* __restrict__ out) {
  // k-major LDS tiles, double buffered: xs[buf][k][b], ws[buf][k][o] (64KB total)
  __shared__ __align__(16) float xs[2][BK * BM];
  __shared__ __align__(16) float ws[2][BK * BN];

  const int tid = threadIdx.x;
  const int n0 = blockIdx.x * BN;   // gridDim.x = N_DIM/BN = 8
  const int b0 = blockIdx.y * BM;   // gridDim.y = B_DIM/BM = 16

  const int tx = tid & 15;          // 16 x TN = 64 = BN
  const int ty = tid >> 4;          // 16 x TM = 64 = BM

  // Per-thread staging pattern: e = tid + t*256 over BM*BK = 4096 elems;
  // row = e/BK, kk = e%BK. Lanes walk kk => coalesced 128B global reads.
  auto stage = [&](int c, int buf) {
#pragma unroll
    for (int t = 0; t < (BM * BK) / NTHREADS; ++t) {
      const int e   = tid + t * NTHREADS;
      const int row = e >> 6;        // / BK
      const int kk  = e & (BK - 1);
      const int k0  = c * BK;
      async_copy_b32(&x[(size_t)(b0 + row) * K_DIM + k0 + kk], &xs[buf][kk * BM + row]);
      async_copy_b32(&W[(size_t)(n0 + row) * K_DIM + k0 + kk], &ws[buf][kk * BN + row]);
    }
  };

  float acc[TM][TN];
#pragma unroll
  for (int i = 0; i < TM; ++i)
#pragma unroll
    for (int j = 0; j < TN; ++j) acc[i][j] = -__builtin_inff();

  stage(0, 0);

#pragma unroll 1
  for (int c = 0; c < NCHUNK; ++c) {
    const int buf = c & 1;
    if (c + 1 < NCHUNK) {
      stage(c + 1, buf ^ 1);     // prefetch next chunk into other buffer
      wait_async<32>();          // in-order completion => chunk c fully landed
    } else {
      wait_async<0>();
    }
    __syncthreads();             // all waves' chunk-c staging visible

    const float* xb = &xs[buf][ty * TM];
    const float* wb = &ws[buf][tx * TN];
    // Per k-step: one b128 of 4 b-values + one b128 of 4 o-values.
    // k unrolled by 2 so each acc update is fmax(acc, fmax(t0,t1)) -> v_max3.
#pragma unroll 4
    for (int kk = 0; kk < BK; kk += 2) {
      const float4 xv0 = *(const float4*)&xb[(kk + 0) * BM];
      const float4 xv1 = *(const float4*)&xb[(kk + 1) * BM];
      const float4 wv0 = *(const float4*)&wb[(kk + 0) * BN];
      const float4 wv1 = *(const float4*)&wb[(kk + 1) * BN];
      const float xa0[TM] = {xv0.x, xv0.y, xv0.z, xv0.w};
      const float xa1[TM] = {xv1.x, xv1.y, xv1.z, xv1.w};
      const float wa0[TN] = {wv0.x, wv0.y, wv0.z, wv0.w};
      const float wa1[TN] = {wv1.x, wv1.y, wv1.z, wv1.w};
#pragma unroll
      for (int i = 0; i < TM; ++i) {
#pragma unroll
        for (int j = 0; j < TN; ++j) {
          const float t0 = xa0[i] + wa0[j];
          const float t1 = xa1[i] + wa1[j];
          acc[i][j] = fmaxf(acc[i][j], fmaxf(t0, t1));  // -> v_max3_num_f32
        }
      }
    }
    __syncthreads();             // buffer buf is reused for chunk c+2
  }

  // Write 4x4 tile; lanes tx=0..15 store consecutive float4 => coalesced.
#pragma unroll
  for (int i = 0; i < TM; ++i) {
    float4 o4;
    o4.x = acc[i][0]; o4.y = acc[i][1]; o4.z = acc[i][2]; o4.w = acc[i][3];
    *(float4*)&out[(size_t)(b0 + ty * TM + i) * N_DIM + n0 + tx * TN] = o4;
  }
}

extern "C" void kernel_launch(void* const* d_in, const int* in_sizes, int n_in,
                              void* d_out, int out_size, void* d_ws, size_t ws_size,
                              hipStream_t stream) {
  (void)in_sizes; (void)n_in; (void)out_size; (void)d_ws; (void)ws_size;
  const float* x = (const float*)d_in[0];   // [1024, 512]
  const float* W = (const float*)d_in[1];   // [512, 512]
  float* out = (float*)d_out;               // [1024, 512]
  dim3 grid(N_DIM / BN, B_DIM / BM);        // (8, 16)
  TropicalLinear_66116726555066_kernel<<<grid, NTHREADS, 0, stream>>>(x, W, out);
}